// _LightweightConv1d_21646635172130
// MI455X (gfx1250) — compile-verified
//
#include <hip/hip_runtime.h>
#include <hip/hip_bf16.h>

typedef __attribute__((ext_vector_type(2))) float v2f;
typedef __attribute__((ext_vector_type(8))) float v8f;

#define T_DIM   4096
#define C_DIM   1024
#define B_DIM   16
#define HEADS   16
#define KTAPS   31
#define PAD     15
#define TBLK    512
#define STRIDE  549          // odd, 549 % 64 = 37; conflict-free transposed reads
#define NROW    16

// ---------------------------------------------------------------------------
// Setup kernel: per-head softmax of the 31 taps, expanded into the Toeplitz
// WMMA A-fragments. Layout: atab[((h*32 + lane)*12 + kk)*2 + {0,1}]
//   lane<16 : VGPR0=A[m,4kk+0], VGPR1=A[m,4kk+1]
//   lane>=16: VGPR0=A[m,4kk+2], VGPR1=A[m,4kk+3]      (m = lane & 15)
//   A[m,j] = softmax(w_h)[j-m] for 0 <= j-m <= 30 else 0
// ---------------------------------------------------------------------------
__global__ void build_atable(const float* __restrict__ weight,
                             float* __restrict__ atab) {
    int t = blockIdx.x * blockDim.x + threadIdx.x;     // 0..511
    if (t >= HEADS * 32) return;
    int h    = t >> 5;
    int lane = t & 31;
    const float* w = weight + h * KTAPS;
    float mx = w[0];
    for (int k = 1; k < KTAPS; ++k) mx = fmaxf(mx, w[k]);
    float s = 0.f;
    for (int k = 0; k < KTAPS; ++k) s += expf(w[k] - mx);
    float inv = 1.f / s;
    int m   = lane & 15;
    int dlt = (lane < 16) ? 0 : 2;
    for (int kk = 0; kk < 12; ++kk) {
        int j0 = 4 * kk + dlt;
        int i0 = j0 - m;
        int i1 = j0 + 1 - m;
        float a0 = (i0 >= 0 && i0 <= 30) ? expf(w[i0] - mx) * inv : 0.f;
        float a1 = (i1 >= 0 && i1 <= 30) ? expf(w[i1] - mx) * inv : 0.f;
        float* p = atab + ((size_t)(h * 32 + lane) * 12 + kk) * 2;
        p[0] = a0;
        p[1] = a1;
    }
}

// ---------------------------------------------------------------------------
// Main conv kernel. Grid: x = T/TBLK (8), y = head (16), z = b*4+rg (64).
// Workgroup: 256 threads = 8 waves. Each workgroup: 16 channels (sharing one
// head, same batch) x TBLK time. Each wave computes 4 of the 32 16x16 tiles
// with 12 chained V_WMMA_F32_16X16X4_F32 each.
// ---------------------------------------------------------------------------
__global__ __launch_bounds__(256, 1)
void conv_kernel(const float* __restrict__ x,
                 const float* __restrict__ bias,
                 const float* __restrict__ atab,
                 float* __restrict__ out) {
    __shared__ float lds[NROW * STRIDE];               // 35,136 B

    const int tblk = blockIdx.x;
    const int h    = blockIdx.y;
    const int z    = blockIdx.z;
    const int b    = z >> 2;
    const int rg   = z & 3;
    const int t0   = tblk * TBLK;
    const int tid  = threadIdx.x;

    // ---- Phase 1: stage x tile into LDS (coalesced float4, zero-padded) ----
    // u in [0, 548), t = t0 - 16 + u, rows n = 0..15, c = h + 256*rg + 16*n
    const int NV4 = 137;                               // 548 / 4 float4 per row
    for (int idx = tid; idx < NROW * NV4; idx += 256) {
        int n = idx / NV4;
        int q = idx - n * NV4;
        int u = q << 2;
        int t = t0 - 16 + u;
        int c = h + (rg << 8) + (n << 4);
        const float* src = x + ((size_t)b * C_DIM + c) * T_DIM;
        float4 v;
        if (t >= 0 && t + 3 < T_DIM) {
            v = *(const float4*)(src + t);
        } else {
            v.x = (t + 0 >= 0 && t + 0 < T_DIM) ? src[t + 0] : 0.f;
            v.y = (t + 1 >= 0 && t + 1 < T_DIM) ? src[t + 1] : 0.f;
            v.z = (t + 2 >= 0 && t + 2 < T_DIM) ? src[t + 2] : 0.f;
            v.w = (t + 3 >= 0 && t + 3 < T_DIM) ? src[t + 3] : 0.f;
        }
        float* dst = lds + n * STRIDE + u;
        dst[0] = v.x; dst[1] = v.y; dst[2] = v.z; dst[3] = v.w;
    }
    __syncthreads();

    // ---- Phase 2: WMMA tiles ----
    const int wv   = tid >> 5;                         // wave 0..7
    const int lane = tid & 31;
    const int nn   = lane & 15;                        // N index (channel row)
    const int dlt  = (lane < 16) ? 0 : 2;              // half-wave K offset

    // Load precomputed Toeplitz A fragments (12 x float2 per lane)
    v2f afr[12];
    const float* ab = atab + (size_t)(h * 32 + lane) * 12 * 2;
    #pragma unroll
    for (int kk = 0; kk < 12; ++kk)
        afr[kk] = *(const v2f*)(ab + kk * 2);

    const float bv = bias[h + (rg << 8) + (nn << 4)];  // fold bias into acc

    v8f accs[4];
    #pragma unroll
    for (int cc = 0; cc < 4; ++cc) {
        const int ch = wv + (cc << 3);                 // chunk 0..31
        v8f acc = {bv, bv, bv, bv, bv, bv, bv, bv};
        // B[j, nn] = lds[nn*STRIDE + 16*ch + 1 + j], j = 4kk + dlt + {0,1}
        const float* brow = lds + nn * STRIDE + (ch << 4) + 1 + dlt;
        #pragma unroll
        for (int kk = 0; kk < 12; ++kk) {
            v2f bfr;
            bfr.x = brow[4 * kk + 0];
            bfr.y = brow[4 * kk + 1];
            acc = __builtin_amdgcn_wmma_f32_16x16x4_f32(
                /*neg_a=*/false, afr[kk],
                /*neg_b=*/false, bfr,
                /*c_mod=*/(short)0, acc,
                /*reuse_a=*/false, /*reuse_b=*/false);
        }
        accs[cc] = acc;
    }
    __syncthreads();                                   // x tile fully consumed

    // ---- Transpose C through LDS (reuse buffer) ----
    const int ibase = (lane < 16) ? 0 : 8;             // M = r + ibase
    #pragma unroll
    for (int cc = 0; cc < 4; ++cc) {
        const int ch = wv + (cc << 3);
        float* drow = lds + nn * STRIDE + (ch << 4) + ibase;
        #pragma unroll
        for (int r = 0; r < 8; ++r)
            drow[r] = accs[cc][r];
    }
    __syncthreads();

    // ---- Phase 3: coalesced float4 stores ----
    const int row  = tid >> 4;                         // 0..15 channel rows
    const int colb = (tid & 15) << 2;
    const int c    = h + (rg << 8) + (row << 4);
    float* dst = out + ((size_t)b * C_DIM + c) * T_DIM + t0;
    const float* srow = lds + row * STRIDE;
    #pragma unroll
    for (int it = 0; it < 8; ++it) {
        int col = colb + (it << 6);
        float4 v;
        v.x = srow[col + 0];
        v.y = srow[col + 1];
        v.z = srow[col + 2];
        v.w = srow[col + 3];
        *(float4*)(dst + col) = v;
    }
}

// ---------------------------------------------------------------------------
extern "C" void kernel_launch(void* const* d_in, const int* in_sizes, int n_in,
                              void* d_out, int out_size, void* d_ws, size_t ws_size,
                              hipStream_t stream) {
    const float* x      = (const float*)d_in[0];   // (16, 1024, 4096)
    const float* weight = (const float*)d_in[1];   // (16, 1, 31)
    const float* bias   = (const float*)d_in[2];   // (1024,)
    float* out  = (float*)d_out;                   // (16, 1024, 4096)
    float* atab = (float*)d_ws;                    // 16*32*12*2 floats = 48 KB

    build_atable<<<1, 512, 0, stream>>>(weight, atab);

    dim3 grid(T_DIM / TBLK, HEADS, B_DIM * 4);     // (8, 16, 64)
    conv_kernel<<<grid, 256, 0, stream>>>(x, bias, atab, out);
}